// FeatureNet_5652176962284
// MI455X (gfx1250) — compile-verified
//
#include <hip/hip_runtime.h>

typedef float v2f __attribute__((ext_vector_type(2)));
typedef float v8f __attribute__((ext_vector_type(8)));

#define NT 8   // 8 column tiles of 16 -> 128 hidden slots (120 used, 8 zero-padded)
#define KS 6   // 6 K-steps of 4 -> K=24 (22 features + ones-column for bias + zero pad)

// Column n of the implicit 24x128 h-weight matrix:
//   row I[n] -> w0, row J[n] -> w1, row 22 -> bias (A supplies 1.0 there), row 23 -> 0
__device__ __forceinline__ float wh_entry(int kk, int In, int Jn,
                                          float w0, float w1, float hbn) {
    float v = 0.0f;
    if (kk == In)      v = w0;
    else if (kk == Jn) v = w1;
    else if (kk == 22) v = hbn;
    return v;
}

__global__ __launch_bounds__(256) void featurenet_wmma_kernel(
    const float* __restrict__ x,    // (B,25)
    const float* __restrict__ hw,   // (120,2)
    const float* __restrict__ hb,   // (120)
    const float* __restrict__ gW,   // (120,3)
    const float* __restrict__ gb,   // (120)
    const float* __restrict__ bW,   // (120,3)
    const float* __restrict__ bb,   // (120)
    const float* __restrict__ oW,   // (120)
    const float* __restrict__ ob,   // (1)
    float* __restrict__ out,        // (B)
    int B)
{
    // ---- block-shared weight fragments (identical for all 8 waves -> LDS, not VGPRs)
    __shared__ v2f   whB_lds[NT][KS][32];   // 12 KB
    __shared__ v2f   gB_lds[NT][32];        //  2 KB
    __shared__ v2f   bB_lds[NT][32];        //  2 KB
    __shared__ float ow_lds[NT][16];        // 0.5 KB

    const int tid  = threadIdx.x;
    const int lane = tid & 31;
    const int ln   = lane & 15;   // col-within-tile (B/C frags), row-within-tile (A frag)
    const int half = lane >> 4;   // K-half for A/B frags, row+8 for C frags

    // One-time setup: 256 threads == 8 tiles x 32 lane-slots, one fragment column each.
    {
        const int t    = tid >> 5;
        const int slot = tid & 31;
        const int sln  = slot & 15;
        const int shf  = slot >> 4;
        const int n = t * 16 + sln;
        const bool valid = (n < 120);
        const float w0  = valid ? hw[2*n]   : 0.0f;
        const float w1  = valid ? hw[2*n+1] : 0.0f;
        const float hbn = valid ? hb[n]     : 0.0f;
        // GROUP1 = {0..5,18..21}: I[n] = n/12 (<6) else n/12+12 ; GROUP2: J[n] = 6 + n%12
        const int In = valid ? ((n / 12 < 6) ? (n / 12) : (n / 12 + 12)) : -1;
        const int Jn = valid ? (6 + n % 12) : -1;
        #pragma unroll
        for (int s = 0; s < KS; ++s) {
            const int k0 = 4 * s + 2 * shf;  // lanes 0-15: K=4s,4s+1 ; lanes 16-31: K=4s+2,4s+3
            v2f w; w.x = wh_entry(k0,     In, Jn, w0, w1, hbn);
                   w.y = wh_entry(k0 + 1, In, Jn, w0, w1, hbn);
            whB_lds[t][s][slot] = w;
        }
        v2f g, be;
        if (!shf) {   // K=0,1 : W[:,0], W[:,1]
            g.x  = valid ? gW[3*n]   : 0.0f;  g.y  = valid ? gW[3*n+1] : 0.0f;
            be.x = valid ? bW[3*n]   : 0.0f;  be.y = valid ? bW[3*n+1] : 0.0f;
        } else {      // K=2,3 : W[:,2], bias row (A supplies 1.0)
            g.x  = valid ? gW[3*n+2] : 0.0f;  g.y  = valid ? gb[n] : 0.0f;
            be.x = valid ? bW[3*n+2] : 0.0f;  be.y = valid ? bb[n] : 0.0f;
        }
        gB_lds[t][slot] = g;
        bB_lds[t][slot] = be;
        if (!shf) ow_lds[t][sln] = valid ? oW[n] : 0.0f;
    }
    __syncthreads();

    const float outb = ob[0];
    const int wave   = (blockIdx.x * blockDim.x + threadIdx.x) >> 5;
    const int nwaves = (gridDim.x * blockDim.x) >> 5;

    // ---------------- main loop: one 16-row tile per wave iteration
    const int ntiles = B >> 4;
    for (int tile = wave; tile < ntiles; tile += nwaves) {
        const int row0 = tile * 16;
        const float* xr = x + (size_t)(row0 + ln) * 25;   // A frag: all lanes row M=ln

        // prefetch this wave's next row tile (grid-stride) into the caches
        if (tile + nwaves < ntiles)
            __builtin_prefetch(x + (size_t)((tile + nwaves) * 16 + ln) * 25, 0, 1);

        // A fragments for the h GEMM (K=24)
        v2f hA[KS];
        #pragma unroll
        for (int s = 0; s < 5; ++s) {
            const int k0 = 4 * s + 2 * half;
            hA[s].x = xr[k0];
            hA[s].y = xr[k0 + 1];
        }
        if (!half) { hA[5].x = xr[20]; hA[5].y = xr[21]; }  // K=20,21 (features)
        else       { hA[5].x = 1.0f;   hA[5].y = 0.0f;   }  // K=22 (bias one), K=23 (pad)

        // A fragment for gamma/beta GEMMs (K=4 = cond0,cond1,cond2,1)
        v2f cA;
        if (!half) { cA.x = xr[22]; cA.y = xr[23]; }
        else       { cA.x = xr[24]; cA.y = 1.0f;   }

        float acc[8];
        #pragma unroll
        for (int e = 0; e < 8; ++e) acc[e] = 0.0f;

        #pragma unroll 2
        for (int t = 0; t < NT; ++t) {
            // fragment loads from LDS (lane-contiguous, bank-conflict-free)
            v2f wB0 = whB_lds[t][0][lane];
            v2f wB1 = whB_lds[t][1][lane];
            v2f wB2 = whB_lds[t][2][lane];
            v2f wB3 = whB_lds[t][3][lane];
            v2f wB4 = whB_lds[t][4][lane];
            v2f wB5 = whB_lds[t][5][lane];
            v2f gB  = gB_lds[t][lane];
            v2f bB  = bB_lds[t][lane];
            float owt = ow_lds[t][ln];

            v8f hC = {};
            hC = __builtin_amdgcn_wmma_f32_16x16x4_f32(false, hA[0], false, wB0, (short)0, hC, false, false);
            hC = __builtin_amdgcn_wmma_f32_16x16x4_f32(false, hA[1], false, wB1, (short)0, hC, false, false);
            hC = __builtin_amdgcn_wmma_f32_16x16x4_f32(false, hA[2], false, wB2, (short)0, hC, false, false);
            hC = __builtin_amdgcn_wmma_f32_16x16x4_f32(false, hA[3], false, wB3, (short)0, hC, false, false);
            hC = __builtin_amdgcn_wmma_f32_16x16x4_f32(false, hA[4], false, wB4, (short)0, hC, false, false);
            hC = __builtin_amdgcn_wmma_f32_16x16x4_f32(false, hA[5], false, wB5, (short)0, hC, false, false);
            v8f gC = {};
            gC = __builtin_amdgcn_wmma_f32_16x16x4_f32(false, cA, false, gB, (short)0, gC, false, false);
            v8f bC = {};
            bC = __builtin_amdgcn_wmma_f32_16x16x4_f32(false, cA, false, bB, (short)0, bC, false, false);

            #pragma unroll
            for (int e = 0; e < 8; ++e) {
                float tv = fmaxf(gC[e] * hC[e] + bC[e], 0.0f);  // relu(gamma*h + beta)
                acc[e] = fmaf(tv, owt, acc[e]);                 // * out_W[n], sum over n
            }
        }

        // reduce the 16 lanes of each half (sum over hidden cols), write rows
        #pragma unroll
        for (int e = 0; e < 8; ++e) {
            float v = acc[e];
            v += __shfl_xor(v, 1, 32);
            v += __shfl_xor(v, 2, 32);
            v += __shfl_xor(v, 4, 32);
            v += __shfl_xor(v, 8, 32);
            if (ln == 0) out[row0 + half * 8 + e] = v + outb;   // row M = e + 8*half
        }
    }

    // ---------------- remainder rows (B % 16): plain scalar path
    const int rem0 = ntiles << 4;
    const int gtid = blockIdx.x * blockDim.x + threadIdx.x;
    for (int row = rem0 + gtid; row < B; row += gridDim.x * blockDim.x) {
        const float* xr = x + (size_t)row * 25;
        const float c0 = xr[22], c1 = xr[23], c2 = xr[24];
        float o = 0.0f;
        for (int n = 0; n < 120; ++n) {
            const int In = (n / 12 < 6) ? (n / 12) : (n / 12 + 12);
            const int Jn = 6 + n % 12;
            const float h  = hw[2*n] * xr[In] + hw[2*n+1] * xr[Jn] + hb[n];
            const float g  = gW[3*n] * c0 + gW[3*n+1] * c1 + gW[3*n+2] * c2 + gb[n];
            const float be = bW[3*n] * c0 + bW[3*n+1] * c1 + bW[3*n+2] * c2 + bb[n];
            o = fmaf(fmaxf(g * h + be, 0.0f), oW[n], o);
        }
        out[row] = o + ob[0];
    }
}

extern "C" void kernel_launch(void* const* d_in, const int* in_sizes, int n_in,
                              void* d_out, int out_size, void* d_ws, size_t ws_size,
                              hipStream_t stream) {
    const float* x  = (const float*)d_in[0];
    const float* hw = (const float*)d_in[1];
    const float* hb = (const float*)d_in[2];
    const float* gW = (const float*)d_in[3];
    const float* gb = (const float*)d_in[4];
    const float* bW = (const float*)d_in[5];
    const float* bb = (const float*)d_in[6];
    const float* oW = (const float*)d_in[7];
    const float* ob = (const float*)d_in[8];
    float* out = (float*)d_out;

    const int B = in_sizes[0] / 25;   // 524288 in the reference
    // 1024 blocks x 8 waves = 8192 waves -> 4 row-tiles per wave; per-block weight
    // setup (one LDS fill + barrier) is amortized over the tile loop.
    const int blocks = 1024;
    featurenet_wmma_kernel<<<blocks, 256, 0, stream>>>(
        x, hw, hb, gW, gb, bW, bb, oW, ob, out, B);
}